// BktModel_11759620457167
// MI455X (gfx1250) — compile-verified
//
#include <hip/hip_runtime.h>
#include <math.h>

// Problem constants (fixed by the reference)
#define NPROB 10000
#define NKC   100
#define NBATCH 1024
#define NT    500

typedef __attribute__((ext_vector_type(2))) float v2f;
typedef __attribute__((ext_vector_type(8))) float v8f;

__device__ __forceinline__ float lse2(float a, float b) {
  float m = fmaxf(a, b);
  return m + __logf(__expf(a - m) + __expf(b - m));
}

// ---------------------------------------------------------------------------
// Kernel 1: log_t[k,s,o] = trans[k,s,o] - logsumexp_{s'}(trans[k,s',o])
// (log_softmax over axis=1 of [K,2,2]) -> ws, stored TRANSPOSED:
//   logtT[n*NKC + k] with n = s*2+o, so the GEMM's B-fragment reads
//   (k0, k0+1) of one column as a single 8-byte-aligned b64 load.
// ---------------------------------------------------------------------------
__global__ void k_logt(const float* __restrict__ trans, float* __restrict__ logtT) {
  int k = blockIdx.x * blockDim.x + threadIdx.x;
  if (k >= NKC) return;
  const float4 v = reinterpret_cast<const float4*>(trans)[k];
  // elements: v.x=(s0,o0) v.y=(s0,o1) v.z=(s1,o0) v.w=(s1,o1); normalize over s
  const float l0 = lse2(v.x, v.z);
  const float l1 = lse2(v.y, v.w);
  logtT[0 * NKC + k] = v.x - l0;  // n=0 (s=0,o=0)
  logtT[1 * NKC + k] = v.y - l1;  // n=1 (s=0,o=1)
  logtT[2 * NKC + k] = v.z - l0;  // n=2 (s=1,o=0)
  logtT[3 * NKC + k] = v.w - l1;  // n=3 (s=1,o=1)
}

// ---------------------------------------------------------------------------
// Kernel 1b: LO[p,s,o] = log_softmax(obs_logits_problem[p], axis=o)
// Hoists the per-step observation softmax out of the serial scan.
// ---------------------------------------------------------------------------
__global__ void k_lo(const float* __restrict__ obs, float* __restrict__ LO) {
  int p = blockIdx.x * blockDim.x + threadIdx.x;
  if (p >= NPROB) return;
  const float4 v = reinterpret_cast<const float4*>(obs)[p];
  const float ls0 = lse2(v.x, v.y);
  const float ls1 = lse2(v.z, v.w);
  float4 r;
  r.x = v.x - ls0; r.y = v.y - ls0; r.z = v.z - ls1; r.w = v.w - ls1;
  reinterpret_cast<float4*>(LO)[p] = r;
}

// ---------------------------------------------------------------------------
// Kernel 2: packed per-problem table via V_WMMA_F32_16X16X4_F32:
//   M8[p*8 + n]  (n=0..3) = sum_k A[p,k] * log_t[k, s, s2]   (a2b terms)
//   M8[p*8 + 4]          = sum_k A[p,k]                      (sum_chain)
// The rowsum comes free from the WMMA by appending a ones column (N=4) to B.
// GEMM [10000 x 100] x [100 x 5(pad 16)] ; 625 tiles of 16 rows; 25 K-chunks.
// ISA layouts (CDNA5 ISA §7.12.2):
//  A 16x4 f32: lane m=l%16 is row M=m; VGPR0 = K=(hi?2:0), VGPR1 = K=(hi?3:1)
//  B 4x16:     lane n=l%16 is col N=n; VGPR0 = row K=(hi?2:0), VGPR1 = K=(hi?3:1)
//  C/D 16x16:  VGPR v = row M=v (lanes 0-15) / M=v+8 (lanes 16-31), N=lane%16
// B pad is arithmetic (b = bv*nmask + ones4) so the B load stays
// UNCONDITIONAL -> no exec-save/branch sandwiches, just load + pk_fma.
// ---------------------------------------------------------------------------
#if __has_builtin(__builtin_amdgcn_wmma_f32_16x16x4_f32)
#define HAVE_WMMA_F32X4 1
#endif

__global__ void __launch_bounds__(32) k_gemm_M(const float* __restrict__ A,
                                               const float* __restrict__ logtT,
                                               float* __restrict__ M8) {
  const int tile = blockIdx.x;
  const int l = threadIdx.x;
  const int m = l & 15;
  const bool hi = (l >= 16);
#ifdef HAVE_WMMA_F32X4
  const int row = tile * 16 + m;
  const float nmask = (m < 4) ? 1.0f : 0.0f;   // pass-through logt cols 0..3
  const float ones4 = (m == 4) ? 1.0f : 0.0f;  // ones column -> rowsum
  const float* __restrict__ bcolp = logtT + (m & 3) * NKC;
  v8f c = {0.f, 0.f, 0.f, 0.f, 0.f, 0.f, 0.f, 0.f};
#pragma unroll 5
  for (int kk = 0; kk < 25; ++kk) {
    const int k0 = kk * 4 + (hi ? 2 : 0);
    // A fragment: two adjacent floats -> merged b64 load
    const float2 av = *reinterpret_cast<const float2*>(A + row * NKC + k0);
    // B fragment: unconditional b64 load of one logtT column pair,
    // arithmetically masked/augmented (keeps the load unpredicated)
    const float2 bv = *reinterpret_cast<const float2*>(bcolp + k0);
    v2f a, b;
    a.x = av.x;
    a.y = av.y;
    b.x = bv.x * nmask + ones4;
    b.y = bv.y * nmask + ones4;
    c = __builtin_amdgcn_wmma_f32_16x16x4_f32(
        /*neg_a=*/false, a, /*neg_b=*/false, b,
        /*c_mod=*/(short)0, c, /*reuse_a=*/false, /*reuse_b=*/false);
  }
  if (m < 5) {
#pragma unroll
    for (int v = 0; v < 8; ++v) {
      const int r = tile * 16 + v + (hi ? 8 : 0);
      M8[r * 8 + m] = c[v];
    }
  }
#else
  (void)hi;
  if (l < 16) {
    const int row = tile * 16 + m;
    float acc0 = 0.f, acc1 = 0.f, acc2 = 0.f, acc3 = 0.f, acc4 = 0.f;
    for (int k = 0; k < NKC; ++k) {
      const float av = A[row * NKC + k];
      acc0 += av * logtT[0 * NKC + k];
      acc1 += av * logtT[1 * NKC + k];
      acc2 += av * logtT[2 * NKC + k];
      acc3 += av * logtT[3 * NKC + k];
      acc4 += av;
    }
    M8[row * 8 + 0] = acc0; M8[row * 8 + 1] = acc1;
    M8[row * 8 + 2] = acc2; M8[row * 8 + 3] = acc3;
    M8[row * 8 + 4] = acc4;
  }
#endif
}

// ---------------------------------------------------------------------------
// Kernel 3: the sequential HMM scan. One wave32 per batch chain.
// Lane l (l<25) owns KCs k = 4l..4l+3 (chain + 2 alpha states each).
// Per step: 128b gather of chain row, per-lane FMAs, 5-step shfl_xor
// butterfly for {a2[0], a2[1]} only (sum_chain is precomputed in M8 col 4),
// precomputed LO/M8 gathers, tiny uniform scalar math, alpha update,
// 2-float output store. Next step's A row is prefetched
// (global_prefetch_b8) to hide the data-dependent L2 gather latency.
// ---------------------------------------------------------------------------
__global__ void __launch_bounds__(32) k_forward(
    const int* __restrict__ corr, const int* __restrict__ kc,
    const int* __restrict__ problem, const float* __restrict__ A,
    const float* __restrict__ init_logits, const float* __restrict__ M8,
    const float* __restrict__ LO, float* __restrict__ out) {
  const int b = blockIdx.x;
  const int l = threadIdx.x;
  const bool act = (l < 25);

  // init alpha: log_softmax(init_logits, axis=1)
  float al0[4] = {0.f, 0.f, 0.f, 0.f};
  float al1[4] = {0.f, 0.f, 0.f, 0.f};
  if (act) {
    const float4 i0 = reinterpret_cast<const float4*>(init_logits)[2 * l];
    const float4 i1 = reinterpret_cast<const float4*>(init_logits)[2 * l + 1];
    float z;
    z = lse2(i0.x, i0.y); al0[0] = i0.x - z; al1[0] = i0.y - z;
    z = lse2(i0.z, i0.w); al0[1] = i0.z - z; al1[1] = i0.w - z;
    z = lse2(i1.x, i1.y); al0[2] = i1.x - z; al1[2] = i1.y - z;
    z = lse2(i1.z, i1.w); al0[3] = i1.z - z; al1[3] = i1.w - z;
  }

  const int base = b * NT;
  int kci = kc[base];
  for (int t = 0; t < NT; ++t) {
    const int idx = base + t;
    const int pi = problem[idx];
    const int ci = corr[idx];
    // unconditional next-kc load (clamped offset), then prefetch its A row
    const int kcn = kc[idx + ((t + 1 < NT) ? 1 : 0)];
    __builtin_prefetch(A + (size_t)kcn * NKC, 0, 3);

    float4 cj = {0.f, 0.f, 0.f, 0.f};
    if (act) cj = reinterpret_cast<const float4*>(A + (size_t)kci * NKC)[l];

    // per-lane partials: a2[s] = sum_k chain*alpha[.,s]
    float p1 = cj.x * al0[0] + cj.y * al0[1] + cj.z * al0[2] + cj.w * al0[3];
    float p2 = cj.x * al1[0] + cj.y * al1[1] + cj.z * al1[2] + cj.w * al1[3];
#pragma unroll
    for (int msk = 16; msk >= 1; msk >>= 1) {
      p1 += __shfl_xor(p1, msk, 32);
      p2 += __shfl_xor(p2, msk, 32);
    }
    const float a20 = p1, a21 = p2;

    // precomputed per-problem tables (both rows land in single L2 lines)
    const float4 m4 = *reinterpret_cast<const float4*>(M8 + (size_t)kci * 8);
    const float SA = M8[(size_t)kci * 8 + 4];
    const float4 lo = reinterpret_cast<const float4*>(LO)[pi];
    const float LO00 = lo.x, LO01 = lo.y, LO10 = lo.z, LO11 = lo.w;

    // predicted output distribution (normalized)
    const float lp0 = lse2(LO00 * SA + a20, LO10 * SA + a21);
    const float lp1 = lse2(LO01 * SA + a20, LO11 * SA + a21);
    const float lz = lse2(lp0, lp1);

    // alpha update: a3[s] = lse_{s2}( obs_ll[s2] + a2[s2] + a2b[s,s2] )
    const float ol0 = (ci == 0 ? LO00 : LO01) * SA;
    const float ol1 = (ci == 0 ? LO10 : LO11) * SA;
    const float q0 = ol0 + a20;
    const float q1 = ol1 + a21;
    const float t30 = lse2(q0 + m4.x, q1 + m4.y);
    const float t31 = lse2(q0 + m4.z, q1 + m4.w);

    if (act) {
      al0[0] = (1.0f - cj.x) * al0[0] + cj.x * t30;
      al1[0] = (1.0f - cj.x) * al1[0] + cj.x * t31;
      al0[1] = (1.0f - cj.y) * al0[1] + cj.y * t30;
      al1[1] = (1.0f - cj.y) * al1[1] + cj.y * t31;
      al0[2] = (1.0f - cj.z) * al0[2] + cj.z * t30;
      al1[2] = (1.0f - cj.z) * al1[2] + cj.z * t31;
      al0[3] = (1.0f - cj.w) * al0[3] + cj.w * t30;
      al1[3] = (1.0f - cj.w) * al1[3] + cj.w * t31;
    }
    if (l == 0) {
      float2 o2;
      o2.x = lp0 - lz;
      o2.y = lp1 - lz;
      reinterpret_cast<float2*>(out)[idx] = o2;
    }
    kci = kcn;
  }
}

// ---------------------------------------------------------------------------
extern "C" void kernel_launch(void* const* d_in, const int* in_sizes, int n_in,
                              void* d_out, int out_size, void* d_ws, size_t ws_size,
                              hipStream_t stream) {
  (void)in_sizes; (void)n_in; (void)out_size; (void)ws_size;
  const int*   corr    = (const int*)d_in[0];
  const int*   kc      = (const int*)d_in[1];
  const int*   problem = (const int*)d_in[2];
  const float* A       = (const float*)d_in[3];
  const float* trans   = (const float*)d_in[4];
  const float* obs     = (const float*)d_in[5];
  const float* init    = (const float*)d_in[6];
  float* out = (float*)d_out;

  // workspace layout (floats):
  //   [0, 512)            logtT[4][100]
  //   [512, 80512)        M8[10000][8]  (cols 0-3: a2b, col 4: sum_chain)
  //   [80512, 120512)     LO[10000][4]
  float* logtT = (float*)d_ws;
  float* M8    = logtT + 512;
  float* LO    = M8 + (size_t)NPROB * 8;

  k_logt<<<1, 128, 0, stream>>>(trans, logtT);
  k_lo<<<(NPROB + 127) / 128, 128, 0, stream>>>(obs, LO);
  k_gemm_M<<<NPROB / 16, 32, 0, stream>>>(A, logtT, M8);
  k_forward<<<NBATCH, 32, 0, stream>>>(corr, kc, problem, A, init, M8, LO, out);
}